// Net_76330158785143
// MI455X (gfx1250) — compile-verified
//
#include <hip/hip_runtime.h>
#include <hip/hip_bf16.h>

typedef __attribute__((ext_vector_type(2))) float v2f;
typedef __attribute__((ext_vector_type(8))) float v8f;

// im2col offset within one conv2 input channel (12x12), for padded K index rk
// (25 real taps padded to 28). Invalid taps clamp to 0 and are value-masked.
#define BOFF(rk) (((rk) < 25) ? (((rk) / 5) * 12 + ((rk) % 5)) : 0)

// One workgroup (256 threads = 8 waves, wave32) processes one image end-to-end.
// All intermediates live in LDS (~48 KB of 320 KB/WGP). conv2 (85% of FLOPs)
// runs on the fp32 tensor pipe: V_WMMA_F32_16X16X4_F32.
__global__ __launch_bounds__(256) void
fused_cnn_wmma_kernel(const float* __restrict__ x,
                      const float* __restrict__ w1, const float* __restrict__ b1,
                      const float* __restrict__ w2, const float* __restrict__ b2,
                      const float* __restrict__ w3, const float* __restrict__ b3,
                      float* __restrict__ out, int nimg)
{
    __shared__ float s_chp[38 * 38];  // zero-padded input channel (pad 5)
    __shared__ float s_h0[784];       // _cor output, 28x28
    __shared__ float s_h1[32 * 144];  // conv1+relu+pool output, 32x12x12
    __shared__ float s_c2[64 * 64];   // conv2 GEMM result, 64ch x (8x8)
    __shared__ float s_h2[64 * 16];   // pooled conv2, 64x4x4
    __shared__ float s_log[40];       // conv3 outputs, 10 classes x 4 positions
    __shared__ float s_m[10];         // logits

    const int tid = threadIdx.x;
    const int b   = blockIdx.x;
    if (b >= nimg) return;

    // lane/wave decomposition for the WMMA stage (also used for prefetch)
    const int lane   = tid & 31;
    const int wv     = tid >> 5;          // 0..7
    const int laneM  = lane & 15;
    const int laneHi = lane >> 4;         // 0: K={0,1}, 1: K={2,3} (A/B layout)
    const int kap    = laneHi * 2;
    const int m0  = (wv >> 1) * 16;
    const int n0a = (wv & 1) * 32;
    const int n0b = n0a + 16;

    // hint: pull this lane's conv2 weight row toward the caches while we do
    // the cor/conv1 stages (w2 is shared by all 8192 blocks -> L2 resident)
    const float* w2row = w2 + (m0 + laneM) * 800;
    __builtin_prefetch(w2row, 0, 3);
    __builtin_prefetch(w2row + 400, 0, 3);

    // ---------------- stage 0: zero-padded load of channel 2 ----------------
    const float* ch = x + (size_t)b * (3 * 784) + 2 * 784;
    for (int p = tid; p < 38 * 38; p += 256) {
        int i = p / 38 - 5, j = p % 38 - 5;
        float v = 0.f;
        if ((unsigned)i < 28u && (unsigned)j < 28u) v = ch[i * 28 + j];
        s_chp[p] = v;
    }
    __syncthreads();

    // ---------------- stage 1: autocorrelation (11x11 center patch) ---------
    // h0[i][j] = sum_{u,v} chp[i+u][j+v] * chp[13+u][13+v]  (no bounds checks)
    for (int p = tid; p < 784; p += 256) {
        int i = p / 28, j = p % 28;
        const float* ibase = &s_chp[i * 38 + j];
        float acc = 0.f;
        for (int u = 0; u < 11; ++u) {
            const float* rowI = ibase + u * 38;
            const float* rowT = &s_chp[(13 + u) * 38 + 13];
            #pragma unroll
            for (int v = 0; v < 11; ++v)
                acc = fmaf(rowI[v], rowT[v], acc);
        }
        s_h0[p] = acc;
    }
    __syncthreads();

    // ---------------- stage 2: conv1 5x5 VALID (1->32) + relu + maxpool2 ----
    for (int idx = tid; idx < 32 * 144; idx += 256) {
        int c   = idx / 144;
        int rem = idx % 144;
        int py = rem / 12, px = rem % 12;
        const float* wc = w1 + c * 25;
        float bias = b1[c];
        float best = 0.f;                 // relu outputs are >= 0
        #pragma unroll
        for (int dy = 0; dy < 2; ++dy) {
            #pragma unroll
            for (int dx = 0; dx < 2; ++dx) {
                int oy = 2 * py + dy, ox = 2 * px + dx;
                float acc = bias;
                #pragma unroll
                for (int u = 0; u < 5; ++u)
                    #pragma unroll
                    for (int v = 0; v < 5; ++v)
                        acc = fmaf(s_h0[(oy + u) * 28 + ox + v], wc[u * 5 + v], acc);
                best = fmaxf(best, acc);
            }
        }
        s_h1[idx] = best;                 // layout: c*144 + y*12 + x
    }
    __syncthreads();

    // ---------------- stage 3: conv2 as fp32 WMMA GEMM ----------------------
    // C(64x64) = W2(64x800) @ im2col(800x64). K padded 25->28 per channel,
    // 7 k-steps/channel. 16 C-tiles of 16x16; 2 N-adjacent tiles per wave
    // (shared A fragment). Branch-free inner loop: per-lane offsets selected
    // once via cndmask, only the final padded step needs value masking.
    const int y0 = (n0a + laneM) >> 3, x0 = (n0a + laneM) & 7;
    const int y1 = (n0b + laneM) >> 3, x1 = (n0b + laneM) & 7;
    const int base0 = y0 * 12 + x0;
    const int base1 = y1 * 12 + x1;

    // per-lane B-fragment offsets for steps 0..5 (all taps valid there)
    int o0[6], o1[6];
    #pragma unroll
    for (int s = 0; s < 6; ++s) {
        o0[s] = laneHi ? BOFF(4 * s + 2) : BOFF(4 * s);
        o1[s] = laneHi ? BOFF(4 * s + 3) : BOFF(4 * s + 1);
    }

    v8f acc0 = {0.f, 0.f, 0.f, 0.f, 0.f, 0.f, 0.f, 0.f};
    v8f acc1 = acc0;
    const float* aBase = w2row + kap;     // lane's K-pair start within a row

    for (int c = 0; c < 32; ++c) {
        const float* wcc  = aBase + c * 25;
        const float* h1b0 = s_h1 + c * 144 + base0;
        const float* h1b1 = s_h1 + c * 144 + base1;
        #pragma unroll
        for (int s = 0; s < 6; ++s) {
            v2f a, bb0, bb1;
            a.x   = wcc[4 * s];           // two b32 loads: alignment-safe
            a.y   = wcc[4 * s + 1];
            bb0.x = h1b0[o0[s]];
            bb0.y = h1b0[o1[s]];
            bb1.x = h1b1[o0[s]];
            bb1.y = h1b1[o1[s]];
            acc0 = __builtin_amdgcn_wmma_f32_16x16x4_f32(
                       false, a, false, bb0, (short)0, acc0, false, false);
            acc1 = __builtin_amdgcn_wmma_f32_16x16x4_f32(
                       false, a, false, bb1, (short)0, acc1, false, false);
        }
        // step 6: only rk=24 is a real tap, and only on the low lane-half.
        {
            float a24 = wcc[24 - kap];    // in-bounds for both halves
            float bv0 = h1b0[52];         // BOFF(24) = 52
            float bv1 = h1b1[52];
            v2f a, bb0, bb1;
            a.x   = laneHi ? 0.f : a24;  a.y   = 0.f;
            bb0.x = laneHi ? 0.f : bv0;  bb0.y = 0.f;
            bb1.x = laneHi ? 0.f : bv1;  bb1.y = 0.f;
            acc0 = __builtin_amdgcn_wmma_f32_16x16x4_f32(
                       false, a, false, bb0, (short)0, acc0, false, false);
            acc1 = __builtin_amdgcn_wmma_f32_16x16x4_f32(
                       false, a, false, bb1, (short)0, acc1, false, false);
        }
    }

    // write back: bias + relu (C/D layout: M = m0 + v + 8*laneHi, N = n0 + laneM)
    #pragma unroll
    for (int vv = 0; vv < 8; ++vv) {
        int M = m0 + vv + laneHi * 8;
        float bias = b2[M];
        s_c2[M * 64 + n0a + laneM] = fmaxf(acc0[vv] + bias, 0.f);
        s_c2[M * 64 + n0b + laneM] = fmaxf(acc1[vv] + bias, 0.f);
    }
    __syncthreads();

    // ---------------- stage 4: maxpool2 on 8x8 -> s_h2 (64x4x4) -------------
    for (int idx = tid; idx < 1024; idx += 256) {
        int c  = idx >> 4;
        int p  = idx & 15;
        int py = p >> 2, px = p & 3;
        const float* base = &s_c2[c * 64 + py * 16 + px * 2];
        s_h2[idx] = fmaxf(fmaxf(base[0], base[1]), fmaxf(base[8], base[9]));
    }
    __syncthreads();

    // ---------------- stage 5: conv3 3x3 (64->10) + relu --------------------
    if (tid < 40) {
        int o  = tid >> 2;
        int p  = tid & 3;
        int py = p >> 1, px = p & 1;
        const float* wo = w3 + o * 576;
        float acc = b3[o];
        for (int c = 0; c < 64; ++c) {
            const float* hc = &s_h2[c * 16];
            const float* wc = wo + c * 9;
            #pragma unroll
            for (int u = 0; u < 3; ++u)
                #pragma unroll
                for (int v = 0; v < 3; ++v)
                    acc = fmaf(hc[(py + u) * 4 + px + v], wc[u * 3 + v], acc);
        }
        s_log[tid] = fmaxf(acc, 0.f);
    }
    __syncthreads();

    // spatial mean over the 2x2 conv3 outputs
    if (tid < 10)
        s_m[tid] = 0.25f * (s_log[tid * 4] + s_log[tid * 4 + 1] +
                            s_log[tid * 4 + 2] + s_log[tid * 4 + 3]);
    __syncthreads();

    // log_softmax over 10 classes
    if (tid == 0) {
        float mx = s_m[0];
        #pragma unroll
        for (int o = 1; o < 10; ++o) mx = fmaxf(mx, s_m[o]);
        float sum = 0.f;
        #pragma unroll
        for (int o = 0; o < 10; ++o) sum += __expf(s_m[o] - mx);
        float lse = __logf(sum) + mx;
        float* op = out + (size_t)b * 10;
        #pragma unroll
        for (int o = 0; o < 10; ++o) op[o] = s_m[o] - lse;
    }
}

extern "C" void kernel_launch(void* const* d_in, const int* in_sizes, int n_in,
                              void* d_out, int out_size, void* d_ws, size_t ws_size,
                              hipStream_t stream) {
    const float* x  = (const float*)d_in[0];
    const float* w1 = (const float*)d_in[1];
    const float* b1 = (const float*)d_in[2];
    const float* w2 = (const float*)d_in[3];
    const float* b2 = (const float*)d_in[4];
    const float* w3 = (const float*)d_in[5];
    const float* b3 = (const float*)d_in[6];
    float* out = (float*)d_out;

    int nimg = in_sizes[0] / (3 * 28 * 28);   // 8192
    fused_cnn_wmma_kernel<<<nimg, 256, 0, stream>>>(x, w1, b1, w2, b2, w3, b3,
                                                    out, nimg);
}